// AltFilter_78056735638067
// MI455X (gfx1250) — compile-verified
//
#include <hip/hip_runtime.h>
#include <math.h>

typedef __attribute__((ext_vector_type(2))) float v2f;
typedef __attribute__((ext_vector_type(8))) float v8f;

#define THREADS 256
#define NW 8            // waves per block (wave32)
#define G 2             // attention instances per block
#define MR (G * 32)     // rows processed per block
#define Dm 128          // model dim
#define ELEMS 42467328L // 8*16*81*4096 elements per full tensor

__device__ __forceinline__ v8f wmma4(v2f a, v2f b, v8f c) {
    // V_WMMA_F32_16X16X4_F32 : D = A(16x4 f32) * B(4x16 f32) + C(16x16 f32)
    return __builtin_amdgcn_wmma_f32_16x16x4_f32(false, a, false, b, (short)0, c,
                                                 false, false);
}

// D[MRxN] = epi( (A[MRxK] @ B[KxN]) * scale [+ C] ), row-major everywhere.
// A/D/C live in LDS, B in global (L2-resident weights) or LDS.
// Waves own disjoint 16x16 tiles -> wave-uniform control flow (EXEC all ones).
template <bool ADD_C, bool GELU>
__device__ __forceinline__ void gemm_rm(const float* __restrict__ A, int lda,
                                        const float* __restrict__ B, int ldb,
                                        float* __restrict__ Dst, int ldd,
                                        const float* __restrict__ Cadd,
                                        int Mt, int Nt, int K, float scale,
                                        int wave, int l16, int half) {
    const int tot = Mt * Nt;
    for (int tile = wave; tile < tot; tile += NW) {
        int mt = tile / Nt, nt = tile % Nt;
        v8f acc = {};
        const float* Ar = A + (mt * 16 + l16) * lda + 2 * half;
        const float* Bc = B + (2 * half) * ldb + nt * 16 + l16;
        for (int k0 = 0; k0 < K; k0 += 4) {
            v2f a, b;
            a.x = Ar[k0];
            a.y = Ar[k0 + 1];
            b.x = Bc[k0 * ldb];
            b.y = Bc[k0 * ldb + ldb];
            acc = wmma4(a, b, acc);
        }
#pragma unroll
        for (int i = 0; i < 8; ++i) {
            int r  = mt * 16 + half * 8 + i;
            int cc = nt * 16 + l16;
            float v = acc[i] * scale;
            if (ADD_C) v += Cadd[r * ldd + cc];
            if (GELU)  v = 0.5f * v * (1.0f + erff(v * 0.70710678118654752f));
            Dst[r * ldd + cc] = v;
        }
    }
}

__device__ __forceinline__ void layernorm(const float* __restrict__ src,
                                          float* __restrict__ dst,
                                          const float* __restrict__ g,
                                          const float* __restrict__ bta,
                                          int wave, int lane) {
    for (int r = wave; r < MR; r += NW) {
        float x[4];
        float s = 0.f;
#pragma unroll
        for (int j = 0; j < 4; ++j) {
            x[j] = src[r * Dm + lane + 32 * j];
            s += x[j];
        }
#pragma unroll
        for (int o = 16; o > 0; o >>= 1) s += __shfl_xor(s, o, 32);
        float m = s * (1.0f / Dm);
        float vv = 0.f;
#pragma unroll
        for (int j = 0; j < 4; ++j) { float d = x[j] - m; vv += d * d; }
#pragma unroll
        for (int o = 16; o > 0; o >>= 1) vv += __shfl_xor(vv, o, 32);
        float rstd = rsqrtf(vv * (1.0f / Dm) + 1e-5f);
#pragma unroll
        for (int j = 0; j < 4; ++j) {
            int col = lane + 32 * j;
            dst[r * Dm + col] = (x[j] - m) * rstd * g[col] + bta[col];
        }
    }
}

// One block = G attention instances; full transformer fused in LDS.
// PASS==1: tokens along H (patch rows); PASS==2: tokens along Wp (patch cols).
template <int PASS>
__global__ __launch_bounds__(THREADS) void trans_kernel(
    const float* __restrict__ src, const float* __restrict__ W_in,
    const float* __restrict__ ln1_g, const float* __restrict__ ln1_b,
    const float* __restrict__ W_qk, const float* __restrict__ W_v,
    const float* __restrict__ W_out, const float* __restrict__ ln2_g,
    const float* __restrict__ ln2_b, const float* __restrict__ W_ff1,
    const float* __restrict__ W_ff2, const float* __restrict__ W_lin,
    float* __restrict__ dst) {
    extern __shared__ float smem[];
    float* sX = smem;                // MR*64   : input X / attn S / output y
    float* sT = smem + MR * 64;      // MR*128  : t / t2 / t3
    float* sA = sT + MR * Dm;        // MR*128  : tn / val / tn2
    float* sB = sA + MR * Dm;        // MR*128  : q / o / ff-half
    float* sC = sB + MR * Dm;        // MR*128  : k

    const int tid = threadIdx.x;
    const int wave = tid >> 5, lane = tid & 31;
    const int l16 = lane & 15, half = lane >> 4;
    const int n0 = (int)blockIdx.x * G;

    // ---- gather input: X[(inst,t), cin=(c,p,q)] ----
#pragma unroll
    for (int inst = 0; inst < G; ++inst) {
        int n = n0 + inst;
        int wl = n & 31;             // pass1: wp, pass2: hH
        int rest = n >> 5;
        int vv = rest % 9; rest /= 9;
        int uu = rest % 9;
        int bb = rest / 9;
        long base0 = ((long)(bb * 16) * 81 + uu * 9 + vv) * 4096;
        for (int idx = tid; idx < 32 * 64; idx += THREADS) {
            int t = idx >> 6, cin = idx & 63;
            int c = cin >> 2, p = (cin >> 1) & 1, q = cin & 1;
            int hh, ww;
            if (PASS == 1) { hh = 2 * t + p;  ww = 2 * wl + q; }
            else           { hh = 2 * wl + p; ww = 2 * t + q;  }
            sX[inst * 2048 + idx] = src[base0 + (long)c * 331776 + hh * 64 + ww];
        }
    }
    __syncthreads();

    const float scale = 0.08838834764831845f; // 1/sqrt(128)

    // t = X @ W_in                                     (MRx64 @ 64x128)
    gemm_rm<false, false>(sX, 64, W_in, Dm, sT, Dm, nullptr, MR / 16, 8, 64, 1.0f,
                          wave, l16, half);
    __syncthreads();
    // tn = LN1(t)
    layernorm(sT, sA, ln1_g, ln1_b, wave, lane);
    __syncthreads();
    // q = tn @ W_qk[:, :128] * scale ; k = tn @ W_qk[:, 128:] * scale
    gemm_rm<false, false>(sA, Dm, W_qk, 256, sB, Dm, nullptr, MR / 16, 8, Dm,
                          scale, wave, l16, half);
    gemm_rm<false, false>(sA, Dm, W_qk + 128, 256, sC, Dm, nullptr, MR / 16, 8,
                          Dm, scale, wave, l16, half);
    __syncthreads();
    // S = q @ k^T  (per instance, 32x32) -> sX
    for (int gi = wave; gi < G * 4; gi += NW) {
        int inst = gi >> 2, mt = (gi >> 1) & 1, nt = gi & 1;
        v8f acc = {};
        const float* Ar = sB + (inst * 32 + mt * 16 + l16) * Dm + 2 * half;
        const float* Br = sC + (inst * 32 + nt * 16 + l16) * Dm + 2 * half;
        for (int k0 = 0; k0 < Dm; k0 += 4) {
            v2f a, b;
            a.x = Ar[k0]; a.y = Ar[k0 + 1];
            b.x = Br[k0]; b.y = Br[k0 + 1];
            acc = wmma4(a, b, acc);
        }
#pragma unroll
        for (int i = 0; i < 8; ++i)
            sX[(inst * 32 + mt * 16 + half * 8 + i) * 32 + nt * 16 + l16] = acc[i];
    }
    // val = t @ W_v * scale -> sA (tn no longer needed)
    gemm_rm<false, false>(sT, Dm, W_v, Dm, sA, Dm, nullptr, MR / 16, 8, Dm, scale,
                          wave, l16, half);
    __syncthreads();
    // softmax rows of S (32 wide, one lane per element)
    for (int r = wave; r < G * 32; r += NW) {
        float x = sX[r * 32 + lane];
        float mx = x;
#pragma unroll
        for (int o = 16; o > 0; o >>= 1) mx = fmaxf(mx, __shfl_xor(mx, o, 32));
        float e = __expf(x - mx);
        float s = e;
#pragma unroll
        for (int o = 16; o > 0; o >>= 1) s += __shfl_xor(s, o, 32);
        sX[r * 32 + lane] = e / s;
    }
    __syncthreads();
    // o = S @ val  (per instance, 32x32 @ 32x128) -> sB
    for (int gi = wave; gi < G * 16; gi += NW) {
        int inst = gi >> 4, tt = gi & 15, mt = tt >> 3, nt = tt & 7;
        v8f acc = {};
        const float* Ar = sX + (inst * 32 + mt * 16 + l16) * 32 + 2 * half;
        const float* Bc = sA + (inst * 32 + 2 * half) * Dm + nt * 16 + l16;
        for (int k0 = 0; k0 < 32; k0 += 4) {
            v2f a, b;
            a.x = Ar[k0]; a.y = Ar[k0 + 1];
            b.x = Bc[k0 * Dm]; b.y = Bc[k0 * Dm + Dm];
            acc = wmma4(a, b, acc);
        }
#pragma unroll
        for (int i = 0; i < 8; ++i)
            sB[(inst * 32 + mt * 16 + half * 8 + i) * Dm + nt * 16 + l16] = acc[i];
    }
    __syncthreads();
    // t2 = o @ W_out + t -> sT
    gemm_rm<true, false>(sB, Dm, W_out, Dm, sT, Dm, sT, MR / 16, 8, Dm, 1.0f,
                         wave, l16, half);
    __syncthreads();
    // tn2 = LN2(t2) -> sA
    layernorm(sT, sA, ln2_g, ln2_b, wave, lane);
    __syncthreads();
    // FFN split over K: t3 = gelu(tn2@Wff1[:, :128])@Wff2[:128,:]
    //                      + gelu(tn2@Wff1[:,128:])@Wff2[128:,:] + t2
    gemm_rm<false, true>(sA, Dm, W_ff1, 256, sB, Dm, nullptr, MR / 16, 8, Dm,
                         1.0f, wave, l16, half);
    __syncthreads();
    gemm_rm<true, false>(sB, Dm, W_ff2, Dm, sT, Dm, sT, MR / 16, 8, Dm, 1.0f,
                         wave, l16, half);
    __syncthreads();
    gemm_rm<false, true>(sA, Dm, W_ff1 + 128, 256, sB, Dm, nullptr, MR / 16, 8,
                         Dm, 1.0f, wave, l16, half);
    __syncthreads();
    gemm_rm<true, false>(sB, Dm, W_ff2 + 128 * Dm, Dm, sT, Dm, sT, MR / 16, 8, Dm,
                         1.0f, wave, l16, half);
    __syncthreads();
    // y = t3 @ W_lin_out (128->64) -> sX
    gemm_rm<false, false>(sT, Dm, W_lin, 64, sX, 64, nullptr, MR / 16, 4, Dm,
                          1.0f, wave, l16, half);
    __syncthreads();

    // ---- scatter: pass1 -> (b,c,uv,hh,ww); pass2 -> (b,c,uv,ww,hh) ----
#pragma unroll
    for (int inst = 0; inst < G; ++inst) {
        int n = n0 + inst;
        int wl = n & 31;
        int rest = n >> 5;
        int vv = rest % 9; rest /= 9;
        int uu = rest % 9;
        int bb = rest / 9;
        long base0 = ((long)(bb * 16) * 81 + uu * 9 + vv) * 4096;
        for (int idx = tid; idx < 32 * 64; idx += THREADS) {
            int t = idx >> 6, cin = idx & 63;
            int c = cin >> 2, p = (cin >> 1) & 1, q = cin & 1;
            long cb = base0 + (long)c * 331776;
            float v = sX[inst * 2048 + idx];
            if (PASS == 1) dst[cb + (2 * t + p) * 64 + (2 * wl + q)] = v;
            else           dst[cb + (2 * t + q) * 64 + (2 * wl + p)] = v;
        }
    }
}

// Fused 3x(3x3 conv, 16ch, leaky 0.2) + residual, one (b,uv,16x16 tile) per block.
// PASS==2 writes the spatially transposed result into d_out.
template <int PASS>
__global__ __launch_bounds__(THREADS) void conv_kernel(
    const float* __restrict__ src, const float* __restrict__ w1,
    const float* __restrict__ w2, const float* __restrict__ w3,
    const float* __restrict__ resid, float* __restrict__ dst) {
    extern __shared__ float smem[];
    float* sIn = smem;              // 16*22*22
    float* sW  = sIn + 16 * 484;    // 3*2304
    float* sT1 = sW + 3 * 2304;     // 16*20*20
    float* sT2 = sT1 + 16 * 400;    // 16*18*18
    const int tid = threadIdx.x;
    int bx = (int)blockIdx.x;
    int tile = bx & 15, buv = bx >> 4;
    int b = buv / 81, uv = buv % 81;
    int ty = (tile >> 2) * 16, tx = (tile & 3) * 16;
    long sbase = ((long)(b * 16) * 81 + uv) * 4096; // + c*331776 per channel

    for (int i = tid; i < 2304; i += THREADS) {
        sW[i] = w1[i];
        sW[2304 + i] = w2[i];
        sW[4608 + i] = w3[i];
    }
    for (int idx = tid; idx < 16 * 484; idx += THREADS) {
        int c = idx / 484, rem = idx - c * 484;
        int iy = rem / 22, ix = rem - iy * 22;
        int gy = ty + iy - 3, gx = tx + ix - 3;
        float v = 0.f;
        if (gy >= 0 && gy < 64 && gx >= 0 && gx < 64)
            v = src[sbase + (long)c * 331776 + gy * 64 + gx];
        sIn[idx] = v;
    }
    __syncthreads();
    // layer 1: 20x20 region
    for (int idx = tid; idx < 16 * 400; idx += THREADS) {
        int oc = idx / 400, rem = idx - oc * 400;
        int oy = rem / 20, ox = rem - oy * 20;
        float acc = 0.f;
        const float* wr = sW + oc * 144;
        for (int ic = 0; ic < 16; ++ic) {
            const float* ip = sIn + ic * 484 + oy * 22 + ox;
#pragma unroll
            for (int dy = 0; dy < 3; ++dy)
#pragma unroll
                for (int dx = 0; dx < 3; ++dx)
                    acc += ip[dy * 22 + dx] * wr[ic * 9 + dy * 3 + dx];
        }
        sT1[idx] = acc >= 0.f ? acc : 0.2f * acc;
    }
    __syncthreads();
    // layer 2: 18x18 region
    for (int idx = tid; idx < 16 * 324; idx += THREADS) {
        int oc = idx / 324, rem = idx - oc * 324;
        int oy = rem / 18, ox = rem - oy * 18;
        float acc = 0.f;
        const float* wr = sW + 2304 + oc * 144;
        for (int ic = 0; ic < 16; ++ic) {
            const float* ip = sT1 + ic * 400 + oy * 20 + ox;
#pragma unroll
            for (int dy = 0; dy < 3; ++dy)
#pragma unroll
                for (int dx = 0; dx < 3; ++dx)
                    acc += ip[dy * 20 + dx] * wr[ic * 9 + dy * 3 + dx];
        }
        sT2[idx] = acc >= 0.f ? acc : 0.2f * acc;
    }
    __syncthreads();
    // layer 3: 16x16 tile + residual + store
    for (int idx = tid; idx < 4096; idx += THREADS) {
        int oc = idx >> 8, rem = idx & 255;
        int oy = rem >> 4, ox = rem & 15;
        float acc = 0.f;
        const float* wr = sW + 4608 + oc * 144;
        for (int ic = 0; ic < 16; ++ic) {
            const float* ip = sT2 + ic * 324 + oy * 18 + ox;
#pragma unroll
            for (int dy = 0; dy < 3; ++dy)
#pragma unroll
                for (int dx = 0; dx < 3; ++dx)
                    acc += ip[dy * 18 + dx] * wr[ic * 9 + dy * 3 + dx];
        }
        float v = acc >= 0.f ? acc : 0.2f * acc;
        int gy = ty + oy, gx = tx + ox;
        long cb = sbase + (long)oc * 331776;
        long o = (PASS == 1) ? (cb + gy * 64 + gx) : (cb + gx * 64 + gy);
        dst[o] = v + resid[o];
    }
}

extern "C" void kernel_launch(void* const* d_in, const int* in_sizes, int n_in,
                              void* d_out, int out_size, void* d_ws,
                              size_t ws_size, hipStream_t stream) {
    (void)in_sizes; (void)n_in; (void)out_size; (void)ws_size;
    const float* buffer = (const float*)d_in[0];
    const float* W_in   = (const float*)d_in[1];
    const float* ln1_g  = (const float*)d_in[2];
    const float* ln1_b  = (const float*)d_in[3];
    const float* W_qk   = (const float*)d_in[4];
    const float* W_v    = (const float*)d_in[5];
    const float* W_out  = (const float*)d_in[6];
    const float* ln2_g  = (const float*)d_in[7];
    const float* ln2_b  = (const float*)d_in[8];
    const float* W_ff1  = (const float*)d_in[9];
    const float* W_ff2  = (const float*)d_in[10];
    const float* W_lin  = (const float*)d_in[11];
    const float* cw1    = (const float*)d_in[12];
    const float* cw2    = (const float*)d_in[13];
    const float* cw3    = (const float*)d_in[14];
    float* out = (float*)d_out;

    float* A  = (float*)d_ws;       // trans output slab (170 MB)
    float* X1 = A + ELEMS;          // conv-block-1 output slab (170 MB)

    const int trans_blocks = 20736 / G;            // 10368
    const int conv_blocks  = 8 * 81 * 16;          // 10368
    const size_t trans_lds = (size_t)(MR * 64 + 4 * MR * Dm) * sizeof(float); // 147456
    const size_t conv_lds  = (size_t)(16 * 484 + 3 * 2304 + 16 * 400 + 16 * 324)
                             * sizeof(float);      // 104960

    trans_kernel<1><<<trans_blocks, THREADS, trans_lds, stream>>>(
        buffer, W_in, ln1_g, ln1_b, W_qk, W_v, W_out, ln2_g, ln2_b, W_ff1, W_ff2,
        W_lin, A);
    conv_kernel<1><<<conv_blocks, THREADS, conv_lds, stream>>>(A, cw1, cw2, cw3,
                                                               buffer, X1);
    trans_kernel<2><<<trans_blocks, THREADS, trans_lds, stream>>>(
        X1, W_in, ln1_g, ln1_b, W_qk, W_v, W_out, ln2_g, ln2_b, W_ff1, W_ff2,
        W_lin, A);
    conv_kernel<2><<<conv_blocks, THREADS, conv_lds, stream>>>(A, cw1, cw2, cw3,
                                                               buffer, out);
}